// SS_76527727280482
// MI455X (gfx1250) — compile-verified
//
#include <hip/hip_runtime.h>
#include <hip/hip_bf16.h>

typedef __attribute__((ext_vector_type(2))) float v2f;
typedef __attribute__((ext_vector_type(8))) float v8f;

static __device__ __forceinline__ int imin(int a, int b) { return a < b ? a : b; }
static __device__ __forceinline__ int imax(int a, int b) { return a > b ? a : b; }

// ---------------------------------------------------------------------------
// Kernel 1: lengths[b] = sum_s mask[b,s].  One 256-thread block per batch,
// int4-vectorized loads (global_load_b128) to stream the 1MB mask at full BW.
// ---------------------------------------------------------------------------
__global__ void SS_lengths_kernel(const int* __restrict__ mask,
                                  int* __restrict__ lengths, int S) {
    const int b = blockIdx.x;
    const int4* row = (const int4*)(mask + (size_t)b * S);
    const int nvec = S >> 2;  // S divisible by 4 (S=4096)

    int sum = 0;
    for (int i = threadIdx.x; i < nvec; i += blockDim.x) {
        int4 v = row[i];
        sum += v.x + v.y + v.z + v.w;
    }

    __shared__ int sdata[256];
    sdata[threadIdx.x] = sum;
    __syncthreads();
    for (int off = blockDim.x >> 1; off > 0; off >>= 1) {
        if ((int)threadIdx.x < off) sdata[threadIdx.x] += sdata[threadIdx.x + off];
        __syncthreads();
    }
    if (threadIdx.x == 0) lengths[b] = sdata[0];
}

// ---------------------------------------------------------------------------
// Kernel 2: one wave (32 lanes) per (batch b, 16-wide d tile).
// out[b, d0+m] = sum over rows r in [max(len-x,0), min(len,S)) of hidden[b,r,d0+m]
// via chained V_WMMA_F32_16X16X4_F32:
//   A[m,k] = hidden value (16x4 tile, tail lanes zeroed), B = all-ones 4x16,
//   accumulator carried through C.  Exact f32 RNE accumulation.
//
// A layout (32-bit 16x4): lane = (k>=2 ? 16 : 0) + m ; vgpr = k & 1.
// D layout (32-bit 16x16): vgpr j -> lanes 0-15 (M=j, N=lane), lanes 16-31 (M=j+8).
//
// Branchless body: row index is clamped (always in-bounds) so loads are
// unconditional; the window tail mask is a v_cndmask on the loaded value.
// EXEC stays all-1s throughout, as WMMA requires.
// ---------------------------------------------------------------------------
__global__ void SS_window_sum_wmma(const float* __restrict__ hidden,
                                   const int* __restrict__ lengths,
                                   const int* __restrict__ xptr,
                                   float* __restrict__ out,
                                   int S, int D) {
    const int b    = blockIdx.y;
    const int d0   = blockIdx.x * 16;
    const int lane = threadIdx.x & 31;     // wave32
    const int m    = lane & 15;            // d within tile
    const int khi  = (lane >> 4) << 1;     // 0 for lanes 0-15, 2 for lanes 16-31

    const int x     = xptr[0];             // scalar input lives on device
    const int len   = lengths[b];
    const int smax  = imin(len, S);        // valid rows are [start, smax)
    const int start = imax(len - x, 0);
    const int count = smax - start;        // number of weight-1 rows (may be <= 0)

    const float* hb = hidden + (size_t)b * S * D + (size_t)d0 + m;

    v8f acc = {};                          // C/D accumulator
    v2f bones = {1.0f, 1.0f};              // B = all ones (layout-independent)

    for (int base = 0; base < count; base += 4) {
        const int k0 = base + khi;         // 0..count+2
        const int k1 = k0 + 1;
        // Clamped element offsets: always inside [start, smax-1] -> safe loads.
        const int o0 = (start + imin(k0, count - 1)) * D;
        const int o1 = (start + imin(k1, count - 1)) * D;
        const float v0 = hb[o0];
        const float v1 = hb[o1];
        v2f a;
        a[0] = (k0 < count) ? v0 : 0.0f;   // v_cndmask, no exec divergence
        a[1] = (k1 < count) ? v1 : 0.0f;
        // 8 args: (neg_a, A, neg_b, B, c_mod, C, reuse_a, reuse_b)
        acc = __builtin_amdgcn_wmma_f32_16x16x4_f32(
            false, a, false, bones, (short)0, acc, false, false);
    }

    // Every column N of D holds the window sum; lane 0 owns (N=0, M=0..7),
    // lane 16 owns (N=0, M=8..15).  Two lanes emit two b128 stores each.
    if (m == 0) {
        float* dst = out + (size_t)b * D + d0 + ((lane >> 4) << 3);
        float4 lo4 = make_float4(acc[0], acc[1], acc[2], acc[3]);
        float4 hi4 = make_float4(acc[4], acc[5], acc[6], acc[7]);
        ((float4*)dst)[0] = lo4;           // 64B-aligned: d0 multiple of 16
        ((float4*)dst)[1] = hi4;
    }
}

// ---------------------------------------------------------------------------
// Launcher.  Inputs: hidden f32 (B*S*D), mask i32 (B*S), x i32 (1 element).
// Output: f32 (B*D).  Workspace: lengths[B] ints.
// ---------------------------------------------------------------------------
extern "C" void kernel_launch(void* const* d_in, const int* in_sizes, int n_in,
                              void* d_out, int out_size, void* d_ws, size_t ws_size,
                              hipStream_t stream) {
    const float* hidden = (const float*)d_in[0];
    const int*   mask   = (const int*)d_in[1];
    const int*   xptr   = (const int*)d_in[2];
    float*       out    = (float*)d_out;

    const int D = in_sizes[0] / in_sizes[1];   // (B*S*D)/(B*S)
    const int B = out_size / D;                // out is B*1*D
    const int S = in_sizes[1] / B;

    int* lengths = (int*)d_ws;

    SS_lengths_kernel<<<B, 256, 0, stream>>>(mask, lengths, S);

    dim3 grid(D / 16, B);
    SS_window_sum_wmma<<<grid, 32, 0, stream>>>(hidden, lengths, xptr, out, S, D);
}